// SequentialCrossSpectralDensityLayer_pyriemann_76081050681795
// MI455X (gfx1250) — compile-verified
//
#include <hip/hip_runtime.h>

// CDNA5 / gfx1250, wave32.
// One wave per (b, s) frame: 16x256 window -> cov via V_WMMA_F32_16X16X4_F32,
// Jacobi eigensolve in LDS, logm via WMMA, triu extraction.
// Strip staging uses GLOBAL_LOAD_ASYNC_TO_LDS_B128 (ASYNCcnt path).

typedef float v2f __attribute__((ext_vector_type(2)));
typedef float v8f __attribute__((ext_vector_type(8)));

#define Bn     128
#define Cn     16
#define Tn     8192
#define FLEN   256
#define FSTEP  64
#define Sn     125        // (8192-256)/64 + 1
#define SG     4          // frames per block
#define STRIP  448        // (SG-1)*64 + 256
#define STRIDE 452        // padded LDS row stride (floats): 452 % 64 == 4 -> conflict-free b64 frag loads
#define ASTR   17         // padded 16x16 tile stride
#define NSWEEP 8
#define NTRIU  136        // 16*17/2

__global__ __launch_bounds__(128)
void csd_logm_kernel(const float* __restrict__ x, float* __restrict__ out)
{
    __shared__ __align__(16) float strip[Cn * STRIDE];          // shared 16 x 448 window strip
    __shared__ float Atile[SG][Cn * ASTR];                      // per-wave working matrix
    __shared__ float Vtile[SG][Cn * ASTR];                      // per-wave eigenvectors

    const int b    = blockIdx.y;
    const int g    = blockIdx.x;
    const int s0   = g * SG;
    const int tid  = threadIdx.x;
    const int w    = tid >> 5;        // wave id in block
    const int lane = tid & 31;
    const int n    = lane & 15;       // channel / matrix column owned by this lane
    const int h    = lane >> 4;       // half-wave select (K-pair / row-half select)

    // ------------------------------------------------------------------
    // Cooperative strip stage: 16 channels x 448 samples, global -> LDS
    // via async-to-LDS B128 (no VGPR staging, tracked by ASYNCcnt).
    // OOB lanes (only possible in the last s-group, region never read
    // because all its waves clamp to frame 124) are predicated off.
    // ------------------------------------------------------------------
    const float* xb = x + (size_t)b * Cn * Tn;
    const int t0 = s0 * FSTEP;
    for (int i = tid; i < Cn * (STRIP / 4); i += 128) {
        int c  = i / (STRIP / 4);
        int t4 = (i % (STRIP / 4)) * 4;
        int tg = t0 + t4;
        if (tg + 3 < Tn) {
            unsigned lds_addr = (unsigned)(uintptr_t)&strip[c * STRIDE + t4];
            const float* gp = xb + c * Tn + tg;
            asm volatile("global_load_async_to_lds_b128 %0, %1, off"
                         :: "v"(lds_addr), "v"(gp) : "memory");
        }
    }
    asm volatile("s_wait_asynccnt 0x0" ::: "memory");
    __syncthreads();

    // Clamp s so every wave runs identical (EXEC-full) control flow; duplicate
    // waves in the last group redundantly recompute frame 124 (same values).
    const int s   = min(s0 + w, Sn - 1);
    const int off = (s - s0) * FSTEP;          // frame start inside the strip

    // ------------------------------------------------------------------
    // Per-channel mean over this wave's 256-sample window
    // lane: channel n, half h sums samples [h*128, h*128+128)
    // ------------------------------------------------------------------
    float sum = 0.0f;
    {
        const float* rp = &strip[n * STRIDE + off + h * 128];
        #pragma unroll
        for (int u = 0; u < 32; ++u) {
            float4 v = *(const float4*)(rp + 4 * u);
            sum += v.x + v.y + v.z + v.w;
        }
    }
    const float mean = (sum + __shfl_xor(sum, 16)) * (1.0f / 256.0f);

    // ------------------------------------------------------------------
    // Covariance: cov = xc * xc^T over K=256 as 64 x V_WMMA_F32_16X16X4_F32.
    // A-fragment (16x4) and B-fragment (4x16) of xc / xc^T have identical
    // per-lane contents (lane = channel, half selects K pair) -> wmma(a, a, acc).
    // ------------------------------------------------------------------
    v8f acc = {0.f, 0.f, 0.f, 0.f, 0.f, 0.f, 0.f, 0.f};
    {
        const float* fp = &strip[n * STRIDE + off + 2 * h];
        #pragma unroll
        for (int k = 0; k < 64; ++k) {
            v2f a;
            a.x = fp[4 * k + 0] - mean;
            a.y = fp[4 * k + 1] - mean;
            acc = __builtin_amdgcn_wmma_f32_16x16x4_f32(
                false, a, false, a, (short)0, acc, false, false);
        }
    }

    // ------------------------------------------------------------------
    // Stage cov/(FLEN-1) into the per-wave LDS tile, init V = I.
    // C-layout: VGPR j, lane -> element (m = 8*h + j, n).
    // ------------------------------------------------------------------
    volatile float* A = Atile[w];
    volatile float* V = Vtile[w];
    #pragma unroll
    for (int j = 0; j < 8; ++j) {
        int m = 8 * h + j;
        A[m * ASTR + n] = acc[j] * (1.0f / (float)(FLEN - 1));
        V[m * ASTR + n] = (m == n) ? 1.0f : 0.0f;
    }
    __builtin_amdgcn_wave_barrier();

    // ------------------------------------------------------------------
    // Cyclic Jacobi eigensolver (wave-cooperative, per-wave LDS tile).
    // Lanes 0-15: rotate A columns then rows; lanes 16-31: accumulate V.
    // Volatile LDS + in-order DS per wave => no heavy sync needed.
    // ------------------------------------------------------------------
    for (int sweep = 0; sweep < NSWEEP; ++sweep) {
        for (int p = 0; p < Cn - 1; ++p) {
            for (int q = p + 1; q < Cn; ++q) {
                float app = A[p * ASTR + p];
                float aqq = A[q * ASTR + q];
                float apq = A[p * ASTR + q];
                float c_ = 1.0f, s_ = 0.0f;
                if (__builtin_fabsf(apq) > 1e-12f) {     // wave-uniform branch
                    float tau = (aqq - app) / (2.0f * apq);
                    float t_  = ((tau >= 0.0f) ? 1.0f : -1.0f) /
                                (__builtin_fabsf(tau) + __builtin_sqrtf(1.0f + tau * tau));
                    c_ = 1.0f / __builtin_sqrtf(1.0f + t_ * t_);
                    s_ = t_ * c_;
                }
                if (h == 0) {                              // column rotation: A <- A*J
                    float ajp = A[n * ASTR + p], ajq = A[n * ASTR + q];
                    A[n * ASTR + p] = c_ * ajp - s_ * ajq;
                    A[n * ASTR + q] = s_ * ajp + c_ * ajq;
                } else {                                   // eigenvectors: V <- V*J
                    float vjp = V[n * ASTR + p], vjq = V[n * ASTR + q];
                    V[n * ASTR + p] = c_ * vjp - s_ * vjq;
                    V[n * ASTR + q] = s_ * vjp + c_ * vjq;
                }
                __builtin_amdgcn_wave_barrier();
                if (h == 0) {                              // row rotation: A <- J^T*A
                    float apj = A[p * ASTR + n], aqj = A[q * ASTR + n];
                    A[p * ASTR + n] = c_ * apj - s_ * aqj;
                    A[q * ASTR + n] = s_ * apj + c_ * aqj;
                }
                __builtin_amdgcn_wave_barrier();
            }
        }
    }

    // ------------------------------------------------------------------
    // logm = (V * diag(log w)) * V^T : 4 x V_WMMA_F32_16X16X4_F32.
    // Same lane symmetry: a-frag = v*log(lambda), b-frag = v.
    // ------------------------------------------------------------------
    v8f lac = {0.f, 0.f, 0.f, 0.f, 0.f, 0.f, 0.f, 0.f};
    #pragma unroll
    for (int k = 0; k < 4; ++k) {
        int j0 = 4 * k + 2 * h;
        float lam0 = A[j0 * ASTR + j0];
        float lam1 = A[(j0 + 1) * ASTR + (j0 + 1)];
        float l0 = __logf(fmaxf(lam0, 1e-30f));
        float l1 = __logf(fmaxf(lam1, 1e-30f));
        float v0 = V[n * ASTR + j0];
        float v1 = V[n * ASTR + j0 + 1];
        v2f af; af.x = v0 * l0; af.y = v1 * l1;
        v2f bf; bf.x = v0;      bf.y = v1;
        lac = __builtin_amdgcn_wmma_f32_16x16x4_f32(
            false, af, false, bf, (short)0, lac, false, false);
    }

    // ------------------------------------------------------------------
    // Upper-triangle extraction straight from C-layout registers:
    // lane holds column n, VGPR j holds row m = 8*h + j; emit m <= n.
    // triu flat index: base(m) = m*(33-m)/2, k = base(m) + (n - m).
    // ------------------------------------------------------------------
    const int frame = b * Sn + s;
    float* op = out + (size_t)frame * NTRIU;
    #pragma unroll
    for (int j = 0; j < 8; ++j) {
        int m = 8 * h + j;
        if (m <= n) {
            float val = lac[j] * ((m == n) ? 1.0f : 1.41421356237309515f);
            op[(m * (33 - m)) / 2 + (n - m)] = val;
        }
    }
}

extern "C" void kernel_launch(void* const* d_in, const int* in_sizes, int n_in,
                              void* d_out, int out_size, void* d_ws, size_t ws_size,
                              hipStream_t stream)
{
    (void)in_sizes; (void)n_in; (void)out_size; (void)d_ws; (void)ws_size;
    const float* x = (const float*)d_in[0];
    float* out = (float*)d_out;
    dim3 grid((Sn + SG - 1) / SG, Bn);   // 32 x 128 blocks
    dim3 block(128);                     // 4 waves, one frame each
    csd_logm_kernel<<<grid, block, 0, stream>>>(x, out);
}